// Custom_loss_66829691125920
// MI455X (gfx1250) — compile-verified
//
#include <hip/hip_runtime.h>
#include <hip/hip_bf16.h>

typedef __attribute__((ext_vector_type(2))) float v2f;
typedef __attribute__((ext_vector_type(8))) float v8f;

#define NN 96
#define PP 196
#define LL 64
#define DD 128

// One block computes the max-over-P / masked-mean-over-L pooled similarity for
// a single (i, j, stream) triple using the fp32 matrix pipe.
// blockDim = 128 (4 waves). Wave w handles text-token tile [16w, 16w+16).
// Two independent WMMA accumulator chains (even/odd K-steps) keep the matrix
// pipe issuing at rate instead of serializing on accumulate latency.
__global__ __launch_bounds__(128) void simpool_kernel(
    const float* __restrict__ v0, const float* __restrict__ v1,
    const float* __restrict__ tmat, const unsigned char* __restrict__ mask,
    float* __restrict__ out, int paired) {
  const int s = blockIdx.z;
  const int j = blockIdx.x;
  const int i = paired ? j : blockIdx.y;
  const float* v = s ? v1 : v0;

  const int lane = threadIdx.x & 31;
  const int wave = threadIdx.x >> 5;   // L-tile index, 0..3
  const int n    = lane & 15;          // column within 16x16 tile (also A row-in-tile)
  const int hi   = lane >> 4;          // half-wave select (K pairs 0..1 vs 2..3)

  const float* va   = v    + (size_t)i * PP * DD;   // [196][128]
  const float* tb   = tmat + (size_t)j * LL * DD;   // [ 64][128]
  // B fragment base: B[k][n] = T[16*wave + n][4k + k'], k' = 2*hi + {0,1}
  const float* brow = tb + (size_t)(wave * 16 + n) * DD + 2 * hi;

  float colmax = -1e30f;   // running max over all P for column n of this L-tile

  for (int m = 0; m < 13; ++m) {        // 13 M-tiles cover P=196 (pad to 208)
    int r = m * 16 + n;                 // A row for this lane
    if (r > PP - 1) r = PP - 1;         // clamp OOB loads (excluded from max below)
    const float* arow = va + (size_t)r * DD + 2 * hi;

    v8f acc0 = {0.f, 0.f, 0.f, 0.f, 0.f, 0.f, 0.f, 0.f};
    v8f acc1 = {0.f, 0.f, 0.f, 0.f, 0.f, 0.f, 0.f, 0.f};
    #pragma unroll
    for (int k = 0; k < 32; k += 2) {   // K = 128 = 32 steps of 4, 2 chains
      v2f af0; af0.x = arow[4 * k];     af0.y = arow[4 * k + 1];
      v2f bf0; bf0.x = brow[4 * k];     bf0.y = brow[4 * k + 1];
      v2f af1; af1.x = arow[4 * k + 4]; af1.y = arow[4 * k + 5];
      v2f bf1; bf1.x = brow[4 * k + 4]; bf1.y = brow[4 * k + 5];
      acc0 = __builtin_amdgcn_wmma_f32_16x16x4_f32(
          false, af0, false, bf0, (short)0, acc0, false, false);
      acc1 = __builtin_amdgcn_wmma_f32_16x16x4_f32(
          false, af1, false, bf1, (short)0, acc1, false, false);
    }

    // C/D layout: lane(hi,n) VGPR e holds element (M = e + 8*hi, N = n).
    if (m < 12) {
      #pragma unroll
      for (int e = 0; e < 8; ++e) colmax = fmaxf(colmax, acc0[e] + acc1[e]);
    } else if (hi == 0) {
      // tile 12: global rows 192 + e + 8*hi; valid only e<4, hi==0 (P=196)
      #pragma unroll
      for (int e = 0; e < 4; ++e) colmax = fmaxf(colmax, acc0[e] + acc1[e]);
    }
  }

  // merge the two M-halves of the wave -> full max over P per column n
  colmax = fmaxf(colmax, __shfl_xor(colmax, 16, 32));

  // masked word score, then sum over the 16 columns of this wave's L-tile
  float mf = mask[(size_t)j * LL + wave * 16 + n] ? 1.0f : 0.0f;
  float word = colmax * mf;
  word += __shfl_xor(word, 8, 32);
  word += __shfl_xor(word, 4, 32);
  word += __shfl_xor(word, 2, 32);
  word += __shfl_xor(word, 1, 32);

  __shared__ float wpart[4];
  if (lane == 0) wpart[wave] = word;
  __syncthreads();
  if (threadIdx.x == 0) {
    float tot = wpart[0] + wpart[1] + wpart[2] + wpart[3];
    float cnt = 0.f;
    for (int l = 0; l < LL; ++l) cnt += mask[(size_t)j * LL + l] ? 1.0f : 0.0f;
    if (cnt < 1.0f) cnt = 1.0f;
    size_t outIdx = paired ? ((size_t)s * NN + i)
                           : ((size_t)s * NN * NN + (size_t)i * NN + j);
    out[outIdx] = tot / cnt;
  }
}

// Single-block epilogue: symmetric InfoNCE on both 96x96 logit matrices,
// triplet hinge from diagonals + paired-negative sims, weighted sum -> scalar.
__global__ __launch_bounds__(128) void finalize_kernel(
    const float* __restrict__ Lmats,   // [2][96][96] (main, aug)
    const float* __restrict__ negd,    // [2][96]
    const float* __restrict__ ls_ptr,
    float* __restrict__ out) {
  const int t = threadIdx.x;
  float ls = ls_ptr[0];
  ls = fminf(fmaxf(ls, 0.0f), 4.6052f);
  const float sc = expf(ls);
  const float* Lm = Lmats;
  const float* La = Lmats + NN * NN;

  float pm = 0.f, pa = 0.f, tm = 0.f, ta = 0.f;
  if (t < NN) {
    float rmax_m = -1e30f, cmax_m = -1e30f, rmax_a = -1e30f, cmax_a = -1e30f;
    for (int k = 0; k < NN; ++k) {
      rmax_m = fmaxf(rmax_m, Lm[t * NN + k]);
      cmax_m = fmaxf(cmax_m, Lm[k * NN + t]);
      rmax_a = fmaxf(rmax_a, La[t * NN + k]);
      cmax_a = fmaxf(cmax_a, La[k * NN + t]);
    }
    float rs_m = 0.f, cs_m = 0.f, rs_a = 0.f, cs_a = 0.f;
    for (int k = 0; k < NN; ++k) {
      rs_m += expf(sc * (Lm[t * NN + k] - rmax_m));
      cs_m += expf(sc * (Lm[k * NN + t] - cmax_m));
      rs_a += expf(sc * (La[t * NN + k] - rmax_a));
      cs_a += expf(sc * (La[k * NN + t] - cmax_a));
    }
    float lse_rm = sc * rmax_m + logf(rs_m);
    float lse_cm = sc * cmax_m + logf(cs_m);
    float lse_ra = sc * rmax_a + logf(rs_a);
    float lse_ca = sc * cmax_a + logf(cs_a);
    float dm = Lm[t * (NN + 1)];
    float da = La[t * (NN + 1)];
    pm = (lse_rm - sc * dm) + (lse_cm - sc * dm);
    pa = (lse_ra - sc * da) + (lse_ca - sc * da);
    tm = fmaxf(0.0f, 0.7f - dm + negd[t]);
    ta = fmaxf(0.0f, 0.7f - da + negd[NN + t]);
  }

  __shared__ float s0[128], s1[128], s2[128], s3[128];
  s0[t] = pm; s1[t] = pa; s2[t] = tm; s3[t] = ta;
  __syncthreads();
  if (t == 0) {
    float a = 0.f, b = 0.f, c = 0.f, d = 0.f;
    for (int k = 0; k < 128; ++k) { a += s0[k]; b += s1[k]; c += s2[k]; d += s3[k]; }
    float loss_main = a / (NN * 2.0f);          // (mean_row + mean_col)/2
    float loss_aug  = b / (NN * 2.0f);
    float contrast  = 0.5f * (loss_main + loss_aug);
    float trip      = 0.5f * (c / (float)NN + d / (float)NN);
    out[0] = 1.0f * trip + 2.0f * contrast;
  }
}

extern "C" void kernel_launch(void* const* d_in, const int* in_sizes, int n_in,
                              void* d_out, int out_size, void* d_ws, size_t ws_size,
                              hipStream_t stream) {
  const float* v_main = (const float*)d_in[0];
  const float* v_aug  = (const float*)d_in[1];
  const float* t_pos  = (const float*)d_in[2];
  const float* t_neg  = (const float*)d_in[3];
  const unsigned char* m_pos = (const unsigned char*)d_in[4];
  const unsigned char* m_neg = (const unsigned char*)d_in[5];
  const float* lsc    = (const float*)d_in[6];
  float* out = (float*)d_out;

  float* wsf   = (float*)d_ws;
  float* Lmats = wsf;                 // 2*96*96 floats
  float* negd  = wsf + 2 * NN * NN;   // 2*96 floats

  dim3 blk(128, 1, 1);
  // Full NxN pooled-similarity matrices for (v_main,t_pos) and (v_aug,t_pos)
  simpool_kernel<<<dim3(NN, NN, 2), blk, 0, stream>>>(
      v_main, v_aug, t_pos, m_pos, Lmats, 0);
  // Paired (i==j) negative sims with t_neg/m_neg
  simpool_kernel<<<dim3(NN, 1, 2), blk, 0, stream>>>(
      v_main, v_aug, t_neg, m_neg, negd, 1);
  // Scalar loss
  finalize_kernel<<<dim3(1, 1, 1), blk, 0, stream>>>(Lmats, negd, lsc, out);
}